// QuinticKernel_20572893348678
// MI455X (gfx1250) — compile-verified
//
#include <hip/hip_runtime.h>
#include <math.h>

// ---------------------------------------------------------------------------
// QuinticKernel for MI455X (gfx1250, wave32).
// Memory-bound fused elementwise + 3 reductions.  Traffic plan:
//   pass1: read base+noise once -> slice sums (2MB ws) + per-block |kern| partials
//   tiny : reduce partials -> inv = 1/(mean|kern| + 1e-12)
//   pass2: re-read base+noise (largely L2-resident: 192MB L2 vs 206MB set),
//          recompute kern, scale, NT-store output (write-once stream).
// Wave-wide sums use V_WMMA_F32_16X16X4_F32 with B=ones (matrix-unit reduce).
// ---------------------------------------------------------------------------

typedef float v2f __attribute__((ext_vector_type(2)));
typedef float v4f __attribute__((ext_vector_type(4)));
typedef float v8f __attribute__((ext_vector_type(8)));

#define KS 49                 // 7x7 spatial slice
// workspace float-index layout
#define WS_S    0
#define WS_T    1
#define WS_AB   2
#define WS_INV  3
#define WS_PART 4             // per-block partials (<= 2048)
#define WS_SSUM 4096          // per-slice sums (nslices floats)

// Full-wave (32 lane) sum, result broadcast to all lanes.
// A: lane value in VGPR0, 0 in VGPR1 -> A[m,0]=x_m, A[m,2]=x_{m+16}.
// B: all ones (4x16).  D[m,*] = x_m + x_{m+16}.  Each lane sums its 8 D regs
// (= half-wave total), then one shfl_xor(16) completes the reduction.
__device__ __forceinline__ float wave_sum_wmma(float x) {
    v2f a; a[0] = x;    a[1] = 0.0f;
    v2f b; b[0] = 1.0f; b[1] = 1.0f;
    v8f c = {};
    c = __builtin_amdgcn_wmma_f32_16x16x4_f32(
            /*neg_a=*/false, a, /*neg_b=*/false, b,
            /*c_mod=*/(short)0, c, /*reuse_a=*/false, /*reuse_b=*/false);
    float t = ((c[0] + c[1]) + (c[2] + c[3])) + ((c[4] + c[5]) + (c[6] + c[7]));
    t += __shfl_xor(t, 16, 32);
    return t;
}

__device__ __forceinline__ float noise3(float r) {
    r = r * (r + 1e-7f);
    r = r * (r + 1e-7f);
    r = r * (r + 1e-7f);
    return r;
}

// ---------------------------------------------------------------- scalars ---
__global__ void k_scalars(const float* __restrict__ a, const float* __restrict__ b,
                          const float* __restrict__ buf, float* __restrict__ wsf) {
    if (threadIdx.x == 0 && blockIdx.x == 0) {
        float sc = 0.f, sd = 0.f, se = 0.f;
        for (int i = 0; i < 7; ++i) {
            float fi = (float)i;
            sc += sinf(fi + 1.0f);
            sd += 1.0f / (cosf(fi + 1e-9f) + 1e-9f);
            se += sqrtf(fi + 1.0f);
        }
        float S = sc * sd * se;
        float T = 0.f;
        for (int i = 0; i < 23; ++i) {
            int j  = (i * 7919) % 23;
            int kk = (j * 1543) % 23;
            T += buf[(i * 23 + j) * 23 + kk] * 1e-12f;
        }
        wsf[WS_S]  = S;
        wsf[WS_T]  = T;
        wsf[WS_AB] = a[0] * b[0];
    }
}

// ------------------------------------------------------------------ pass 1 ---
// One wave per slice: lane l owns elements l and l+32 of the 49-element slice.
__global__ void k_pass1(const float* __restrict__ base, const float* __restrict__ noise,
                        float* __restrict__ wsf, int nslices) {
    const float S   = wsf[WS_S];
    const float T   = wsf[WS_T];
    const float abS = wsf[WS_AB] * S;
    float* __restrict__ ssum_ws = wsf + WS_SSUM;

    const int lane       = threadIdx.x & 31;
    const int waveInBlk  = threadIdx.x >> 5;
    const int wavesPerBlk = blockDim.x >> 5;
    const int gwave  = blockIdx.x * wavesPerBlk + waveInBlk;
    const int nwaves = gridDim.x * wavesPerBlk;
    const bool hi    = lane < (KS - 32);   // lanes 0..16 own a second element

    float labs = 0.0f;
    for (int s = gwave; s < nslices; s += nwaves) {
        const int e0 = s * KS + lane;
        float v0 = base[e0];
        float v1 = hi ? base[e0 + 32] : 0.0f;

        float ssum = wave_sum_wmma(v0 + v1);          // slice sum, all lanes
        if (lane == 0) ssum_ws[s] = ssum;

        const float add = abS * ssum + T;
        float k0 = v0 * S + add + noise3(noise[e0]);
        labs += fabsf(k0);
        if (hi) {
            float k1 = v1 * S + add + noise3(noise[e0 + 32]);
            labs += fabsf(k1);
        }
    }

    // deterministic block reduction of |kern| partials
    float wsum = wave_sum_wmma(labs);
    __shared__ float red[32];
    if (lane == 0) red[waveInBlk] = wsum;
    __syncthreads();
    if (threadIdx.x == 0) {
        float t = 0.f;
        for (int w = 0; w < wavesPerBlk; ++w) t += red[w];
        wsf[WS_PART + blockIdx.x] = t;
    }
}

// ------------------------------------------------------- finalize the mean ---
__global__ void k_finalize(float* __restrict__ wsf, int nparts, int ntot) {
    __shared__ float red[256];
    float t = 0.f;
    for (int i = threadIdx.x; i < nparts; i += blockDim.x) t += wsf[WS_PART + i];
    red[threadIdx.x] = t;
    __syncthreads();
    for (int off = 128; off > 0; off >>= 1) {
        if ((int)threadIdx.x < off) red[threadIdx.x] += red[threadIdx.x + off];
        __syncthreads();
    }
    if (threadIdx.x == 0) {
        float mean = red[0] / (float)ntot;
        wsf[WS_INV] = 1.0f / (mean + 1e-12f);
    }
}

// ------------------------------------------------------------------ pass 2 ---
// Vectorized b128 stream; base/noise re-reads mostly hit the 192MB L2.
// Output is write-once -> non-temporal store.
__global__ void k_pass2(const float* __restrict__ base, const float* __restrict__ noise,
                        const float* __restrict__ wsf, float* __restrict__ out, int ntot) {
    const float S   = wsf[WS_S];
    const float T   = wsf[WS_T];
    const float abS = wsf[WS_AB] * S;
    const float inv = wsf[WS_INV];
    const float* __restrict__ ssum = wsf + WS_SSUM;

    const int nvec   = ntot >> 2;
    const int tid    = blockIdx.x * blockDim.x + threadIdx.x;
    const int stride = gridDim.x * blockDim.x;

    for (int i = tid; i < nvec; i += stride) {
        const int e = i << 2;
        v4f bv = __builtin_nontemporal_load((const v4f*)(base + e));
        v4f nv = __builtin_nontemporal_load((const v4f*)(noise + e));
        v4f ov;
#pragma unroll
        for (int j = 0; j < 4; ++j) {
            unsigned s = (unsigned)(e + j) / 49u;   // mul-hi const division
            float k = bv[j] * S + abS * ssum[s] + T + noise3(nv[j]);
            ov[j] = k * inv;
        }
        __builtin_nontemporal_store(ov, (v4f*)(out + e));
    }
    // scalar tail (ntot here is divisible by 16, but stay generic)
    const int tail = ntot & 3;
    if (tid < tail) {
        const int e = (ntot & ~3) + tid;
        unsigned s = (unsigned)e / 49u;
        float k = base[e] * S + abS * ssum[s] + T + noise3(noise[e]);
        out[e] = k * inv;
    }
}

// ---------------------------------------------------------------------------
extern "C" void kernel_launch(void* const* d_in, const int* in_sizes, int n_in,
                              void* d_out, int out_size, void* d_ws, size_t ws_size,
                              hipStream_t stream) {
    const float* base  = (const float*)d_in[0];
    const float* a     = (const float*)d_in[1];
    const float* b     = (const float*)d_in[2];
    const float* buf   = (const float*)d_in[3];
    const float* noise = (const float*)d_in[4];
    float* out = (float*)d_out;
    float* wsf = (float*)d_ws;

    const int ntot    = in_sizes[0];       // 1024*512*49
    const int nslices = ntot / KS;         // 524288
    const int NB = 2048, NT = 256;         // 16384 waves, grid-stride

    k_scalars <<<1,  32, 0, stream>>>(a, b, buf, wsf);
    k_pass1   <<<NB, NT, 0, stream>>>(base, noise, wsf, nslices);
    k_finalize<<<1, 256, 0, stream>>>(wsf, NB, ntot);
    k_pass2   <<<NB, NT, 0, stream>>>(base, noise, wsf, out, ntot);
}